// HierarchicalAttention_49022756716582
// MI455X (gfx1250) — compile-verified
//
#include <hip/hip_runtime.h>
#include <hip/hip_bf16.h>

// Problem constants from the reference: P=8192, C=16384, H=64.
#define P_N 8192
#define C_N 16384
#define HID 64

typedef __attribute__((ext_vector_type(2))) float v2f;
typedef __attribute__((ext_vector_type(8))) float v8f;

// ---- order-preserving float<->uint key (for deterministic atomicMax) ----
__device__ __forceinline__ unsigned f2key(float x) {
    unsigned b = __float_as_uint(x);
    return (b & 0x80000000u) ? ~b : (b | 0x80000000u);
}
__device__ __forceinline__ float key2f(unsigned k) {
    unsigned b = (k & 0x80000000u) ? (k ^ 0x80000000u) : ~k;
    return __uint_as_float(b);
}

// Workspace layout (float index units):
//   [0]              maxkey (as unsigned)
//   [1]              Z (final)
//   [2 .. 65]        v_unnorm[64] (final)
//   [66 .. 16449]    scores[C_N]
//   [16450 ..16705]  wZ[256]          per-wave Z partials
//   [16706 ..33089]  wV[256*64]       per-wave v partials
#define WS_MAXKEY 0
#define WS_Z      1
#define WS_VUN    2
#define WS_SCORES 66
#define WS_WZ     16450
#define WS_WV     16706
#define NWAVES    256   // kernel2: 32 blocks * 256 threads / 32 lanes

__global__ void ha_init(unsigned* maxkey) {
    if (threadIdx.x == 0) *maxkey = 0u;   // below every valid key
}

// score_c[c] = child[c,:] . w_c ;  global max via monotonic-key atomicMax
__global__ void ha_scores(const float* __restrict__ child,
                          const float* __restrict__ attn_w,
                          float* __restrict__ scores,
                          unsigned* __restrict__ maxkey) {
    __shared__ float s_w[HID];
    __shared__ unsigned red[256];
    int tid = threadIdx.x;
    if (tid < HID) s_w[tid] = attn_w[HID + tid];   // w_c = attn_w[0, H:]
    __syncthreads();

    int c = blockIdx.x * 256 + tid;
    const float4* row = (const float4*)(child + (size_t)c * HID);
    float acc = 0.f;
#pragma unroll
    for (int i = 0; i < 16; ++i) {
        float4 d = row[i];
        acc += d.x * s_w[4 * i + 0] + d.y * s_w[4 * i + 1] +
               d.z * s_w[4 * i + 2] + d.w * s_w[4 * i + 3];
    }
    scores[c] = acc;

    red[tid] = f2key(acc);
    __syncthreads();
    for (int s = 128; s > 0; s >>= 1) {
        if (tid < s) red[tid] = max(red[tid], red[tid + s]);
        __syncthreads();
    }
    if (tid == 0) atomicMax(maxkey, red[0]);
}

// Per-wave: 64 children. Z partial via shfl reduce; v partial via
// V_WMMA_F32_16X16X4_F32: A(16x4)=exp weights replicated over rows,
// B(4x16)=child tile, 4 column-tiles cover H=64. A and B both use the
// k = vgpr + 2*laneHalf convention (permutation-safe pairing).
__global__ void ha_accum(const float* __restrict__ child,
                         const float* __restrict__ scores,
                         const unsigned* __restrict__ maxkey,
                         float* __restrict__ wZ,
                         float* __restrict__ wV) {
    const float m = key2f(*maxkey);
    int gtid = blockIdx.x * 256 + threadIdx.x;
    int wave = gtid >> 5;              // 0..255
    int lane = threadIdx.x & 31;
    int half = lane >> 4;              // 0 or 1
    int n    = lane & 15;
    int base = wave * 64;

    // --- deterministic-per-wave Z partial ---
    float zp = 0.f;
    for (int c = base + lane; c < base + 64; c += 32)
        zp += __expf(scores[c] - m);
#pragma unroll
    for (int off = 16; off > 0; off >>= 1)
        zp += __shfl_xor(zp, off, 32);
    if (lane == 0) wZ[wave] = zp;

    // --- WMMA accumulation over this wave's 64 children, k-steps of 4 ---
    v8f acc0 = {}, acc1 = {}, acc2 = {}, acc3 = {};
    const float2* s2 = (const float2*)scores;
    for (int c0 = base; c0 < base + 64; c0 += 4) {
        int ca = c0 + 2 * half;               // even
        float2 sc = s2[ca >> 1];              // scores[ca], scores[ca+1]
        v2f A;
        A.x = __expf(sc.x - m);
        A.y = __expf(sc.y - m);
        const float* ra = child + (size_t)ca * HID;
        const float* rb = ra + HID;
        v2f B0; B0.x = ra[n +  0]; B0.y = rb[n +  0];
        v2f B1; B1.x = ra[n + 16]; B1.y = rb[n + 16];
        v2f B2; B2.x = ra[n + 32]; B2.y = rb[n + 32];
        v2f B3; B3.x = ra[n + 48]; B3.y = rb[n + 48];
        acc0 = __builtin_amdgcn_wmma_f32_16x16x4_f32(false, A, false, B0, (short)0, acc0, false, false);
        acc1 = __builtin_amdgcn_wmma_f32_16x16x4_f32(false, A, false, B1, (short)0, acc1, false, false);
        acc2 = __builtin_amdgcn_wmma_f32_16x16x4_f32(false, A, false, B2, (short)0, acc2, false, false);
        acc3 = __builtin_amdgcn_wmma_f32_16x16x4_f32(false, A, false, B3, (short)0, acc3, false, false);
    }

    // All D rows are identical; lanes 0..15 of d[0] hold columns n0..n0+15.
    if (half == 0) {
        float* dst = wV + (size_t)wave * HID;
        dst[n +  0] = acc0[0];
        dst[n + 16] = acc1[0];
        dst[n + 32] = acc2[0];
        dst[n + 48] = acc3[0];
    }
}

// Deterministic tree-free reduce of per-wave partials (tiny: 256x65 floats).
__global__ void ha_reduce(const float* __restrict__ wZ,
                          const float* __restrict__ wV,
                          float* __restrict__ Z,
                          float* __restrict__ vun) {
    int h = threadIdx.x;   // 64 threads
    float acc = 0.f;
    for (int w = 0; w < NWAVES; ++w)
        acc += wV[(size_t)w * HID + h];
    vun[h] = acc;
    if (h == 0) {
        float z = 0.f;
        for (int w = 0; w < NWAVES; ++w) z += wZ[w];
        *Z = z;
    }
}

// out[p,:] = vun[:] / Z for all 8192 rows; float4 stores.
__global__ void ha_bcast(const float* __restrict__ vun,
                         const float* __restrict__ Z,
                         float* __restrict__ out) {
    int i = blockIdx.x * 256 + threadIdx.x;   // over P_N*16 float4s
    float invZ = 1.0f / (*Z);
    const float4* v4 = (const float4*)vun;
    float4 val = v4[i & 15];
    val.x *= invZ; val.y *= invZ; val.z *= invZ; val.w *= invZ;
    ((float4*)out)[i] = val;
}

extern "C" void kernel_launch(void* const* d_in, const int* in_sizes, int n_in,
                              void* d_out, int out_size, void* d_ws, size_t ws_size,
                              hipStream_t stream) {
    // Inputs: [0]=parent_embeds (unused: softmax shift-invariance makes the
    // output independent of parents), [1]=child_embeds, [2]=attn_w, [3]=attn_b.
    const float* child  = (const float*)d_in[1];
    const float* attn_w = (const float*)d_in[2];
    float* out = (float*)d_out;

    float*    wsf    = (float*)d_ws;
    unsigned* maxkey = (unsigned*)(wsf + WS_MAXKEY);
    float*    Z      = wsf + WS_Z;
    float*    vun    = wsf + WS_VUN;
    float*    scores = wsf + WS_SCORES;
    float*    wZ     = wsf + WS_WZ;
    float*    wV     = wsf + WS_WV;

    ha_init  <<<1, 64, 0, stream>>>(maxkey);
    ha_scores<<<C_N / 256, 256, 0, stream>>>(child, attn_w, scores, maxkey);
    ha_accum <<<32, 256, 0, stream>>>(child, scores, maxkey, wZ, wV);
    ha_reduce<<<1, HID, 0, stream>>>(wZ, wV, Z, vun);
    ha_bcast <<<(P_N * 16) / 256, 256, 0, stream>>>(vun, Z, out);
}